// TensoRFGrid_1726576856425
// MI455X (gfx1250) — compile-verified
//
#include <hip/hip_runtime.h>
#include <hip/hip_bf16.h>

typedef __attribute__((ext_vector_type(2))) float v2f;
typedef __attribute__((ext_vector_type(4))) float v4f;
typedef __attribute__((ext_vector_type(8))) float v8f;

#define PTS      128      // points per block (= threads per block)
#define RCH      24       // channels per plane
#define COUT     28       // output channels
#define KDIM     72       // 3 * RCH
#define FSTR     76       // padded LDS feature row stride (76*4 = 304 B: 16B-aligned rows)
#define GRID_DIM 256      // X = Y = Z = 256
#define PLANE_MN (GRID_DIM * GRID_DIM)        // 65536
#define PLANE_T_ELEMS (PLANE_MN * RCH)        // 1,572,864 floats
#define VEC_T_ELEMS   (GRID_DIM * RCH)        // 6,144 floats

// ---------------------------------------------------------------------------
// Pass 1: channel-major transpose  [24, M] -> [M, 24]
// Makes each sample point's 24 channels contiguous (96 B, 16 B aligned) so
// the gather phase uses b128 loads hitting ~1.5 cache lines per corner
// instead of 24 lines.
// ---------------------------------------------------------------------------
__global__ __launch_bounds__(256) void transpose_cm_kernel(
    const float* __restrict__ src, float* __restrict__ dst, int M, int total)
{
    int i = blockIdx.x * 256 + threadIdx.x;
    if (i >= total) return;
    int m = i / RCH;            // dest-major index
    int c = i - m * RCH;
    dst[i] = src[c * M + m];    // coalesced write, strided (L2-resident) read
}

// ---------------------------------------------------------------------------
// Bilinear plane sample (align_corners=True, zeros padding) fused with the
// 1D vector sample, channel-major layouts. Writes 24 products to dst[0..23].
// ---------------------------------------------------------------------------
__device__ __forceinline__ void sample_group_t(
    const float* __restrict__ plane_t,   // [H, W, 24]
    const float* __restrict__ vec_t,     // [L, 24]
    float gy, float gx, float gv, float* __restrict__ dst)
{
    const int H = GRID_DIM, W = GRID_DIM, L = GRID_DIM;

    // ---- plane bilinear setup ----
    float iy = (gy + 1.f) * 0.5f * (float)(H - 1);
    float ix = (gx + 1.f) * 0.5f * (float)(W - 1);
    float iy0f = floorf(iy), ix0f = floorf(ix);
    float wy = iy - iy0f,   wx = ix - ix0f;
    int iy0 = (int)iy0f, ix0 = (int)ix0f;
    int iy1 = iy0 + 1,   ix1 = ix0 + 1;

    float w00 = (1.f - wy) * (1.f - wx);
    float w01 = (1.f - wy) * wx;
    float w10 = wy * (1.f - wx);
    float w11 = wy * wx;
    if (iy0 < 0 || iy0 >= H) { w00 = 0.f; w01 = 0.f; }
    if (iy1 < 0 || iy1 >= H) { w10 = 0.f; w11 = 0.f; }
    if (ix0 < 0 || ix0 >= W) { w00 = 0.f; w10 = 0.f; }
    if (ix1 < 0 || ix1 >= W) { w01 = 0.f; w11 = 0.f; }

    int y0 = iy0 < 0 ? 0 : (iy0 > H - 1 ? H - 1 : iy0);
    int y1 = iy1 < 0 ? 0 : (iy1 > H - 1 ? H - 1 : iy1);
    int x0 = ix0 < 0 ? 0 : (ix0 > W - 1 ? W - 1 : ix0);
    int x1 = ix1 < 0 ? 0 : (ix1 > W - 1 ? W - 1 : ix1);

    const float* p00 = plane_t + (y0 * W + x0) * RCH;   // 96 B aligned
    const float* p01 = plane_t + (y0 * W + x1) * RCH;
    const float* p10 = plane_t + (y1 * W + x0) * RCH;
    const float* p11 = plane_t + (y1 * W + x1) * RCH;

    // ---- vector linear setup ----
    float iv = (gv + 1.f) * 0.5f * (float)(L - 1);
    float iv0f = floorf(iv);
    float wv = iv - iv0f;
    int i0 = (int)iv0f, i1 = i0 + 1;
    float wv0 = 1.f - wv, wv1 = wv;
    if (i0 < 0 || i0 >= L) wv0 = 0.f;
    if (i1 < 0 || i1 >= L) wv1 = 0.f;
    int c0 = i0 < 0 ? 0 : (i0 > L - 1 ? L - 1 : i0);
    int c1 = i1 < 0 ? 0 : (i1 > L - 1 ? L - 1 : i1);
    const float* v0 = vec_t + c0 * RCH;
    const float* v1 = vec_t + c1 * RCH;

    // 24 channels = 6 x float4: b128 gathers, b128 LDS stores.
#pragma unroll
    for (int cc = 0; cc < RCH / 4; ++cc) {
        v4f a00 = *(const v4f*)(p00 + cc * 4);
        v4f a01 = *(const v4f*)(p01 + cc * 4);
        v4f a10 = *(const v4f*)(p10 + cc * 4);
        v4f a11 = *(const v4f*)(p11 + cc * 4);
        v4f vv0 = *(const v4f*)(v0 + cc * 4);
        v4f vv1 = *(const v4f*)(v1 + cc * 4);
        v4f s  = a00 * w00 + a01 * w01 + a10 * w10 + a11 * w11;
        v4f vv = vv0 * wv0 + vv1 * wv1;
        *(v4f*)(dst + cc * 4) = s * vv;
    }
}

// ---------------------------------------------------------------------------
// Pass 2: fused gather + feature build + WMMA projection.
// ---------------------------------------------------------------------------
__global__ __launch_bounds__(PTS) void tensorf_fused_kernel(
    const float* __restrict__ xyz,
    const float* __restrict__ xy_t,
    const float* __restrict__ xz_t,
    const float* __restrict__ yz_t,
    const float* __restrict__ x_t,
    const float* __restrict__ y_t,
    const float* __restrict__ z_t,
    const float* __restrict__ f_vec,
    const float* __restrict__ xyz_min,
    const float* __restrict__ xyz_max,
    float* __restrict__ out,
    int Npts)
{
    __shared__ float feat[PTS * FSTR];   // [point][channel] padded
    __shared__ float bmat[32 * FSTR];    // f_vec transposed: [n(0..31)][k] zero-padded

    const int tid = threadIdx.x;
    const int p   = blockIdx.x * PTS + tid;

    // ---- stage B = f_vec^T (zero-padded to N=32) into LDS ----
    for (int i = tid; i < 32 * FSTR; i += PTS) {
        int n = i / FSTR, k = i % FSTR;
        bmat[i] = (n < COUT && k < KDIM) ? f_vec[k * COUT + n] : 0.f;
    }

    // ---- phase A: channel-major gathers -> LDS feature rows ----
    float* row = &feat[tid * FSTR];
    if (p < Npts) {
        float px = xyz[3 * p + 0];
        float py = xyz[3 * p + 1];
        float pz = xyz[3 * p + 2];
        float mn0 = xyz_min[0], mn1 = xyz_min[1], mn2 = xyz_min[2];
        float mx0 = xyz_max[0], mx1 = xyz_max[1], mx2 = xyz_max[2];
        float nx = (px - mn0) / (mx0 - mn0) * 2.f - 1.f;
        float ny = (py - mn1) / (mx1 - mn1) * 2.f - 1.f;
        float nz = (pz - mn2) / (mx2 - mn2) * 2.f - 1.f;

        sample_group_t(xy_t, z_t, nx, ny, nz, row + 0);        // xy_feat * z_feat
        sample_group_t(xz_t, y_t, nx, nz, ny, row + RCH);      // xz_feat * y_feat
        sample_group_t(yz_t, x_t, ny, nz, nx, row + 2 * RCH);  // yz_feat * x_feat
    } else {
        v4f z = {};
#pragma unroll
        for (int c = 0; c < KDIM / 4; ++c) *(v4f*)(row + c * 4) = z;
    }
    __syncthreads();

    // ---- phase B: WMMA GEMM  feat[128x72] @ f_vec[72x28] ----
    const int lane  = tid & 31;
    const int wave  = tid >> 5;
    const int mbase = wave * 32;
    const int mrow  = lane & 15;
    const int halfk = (lane >> 4) << 1;   // lanes 0-15 hold K {0,1}; 16-31 hold K {2,3}

    v8f acc00 = {}, acc01 = {}, acc10 = {}, acc11 = {};

#pragma unroll
    for (int ks = 0; ks < KDIM / 4; ++ks) {
        const int k0 = ks * 4 + halfk;
        v2f a0 = *(const v2f*)&feat[(mbase + mrow) * FSTR + k0];
        v2f a1 = *(const v2f*)&feat[(mbase + 16 + mrow) * FSTR + k0];
        v2f b0 = *(const v2f*)&bmat[mrow * FSTR + k0];
        v2f b1 = *(const v2f*)&bmat[(16 + mrow) * FSTR + k0];
        acc00 = __builtin_amdgcn_wmma_f32_16x16x4_f32(false, a0, false, b0,
                                                      (short)0, acc00, false, false);
        acc01 = __builtin_amdgcn_wmma_f32_16x16x4_f32(false, a0, false, b1,
                                                      (short)0, acc01, false, false);
        acc10 = __builtin_amdgcn_wmma_f32_16x16x4_f32(false, a1, false, b0,
                                                      (short)0, acc10, false, false);
        acc11 = __builtin_amdgcn_wmma_f32_16x16x4_f32(false, a1, false, b1,
                                                      (short)0, acc11, false, false);
    }

    // ---- store: C/D layout VGPR r -> M = r (+8 for lanes 16-31), N = lane&15 ----
    const int gbase = blockIdx.x * PTS + mbase;
    const int moff  = (lane >> 4) << 3;
    const bool fullblk = (blockIdx.x + 1) * PTS <= Npts;
    int n0 = mrow;
    int n1 = 16 + mrow;
    if (fullblk) {
#pragma unroll
        for (int r = 0; r < 8; ++r) {
            int m0 = gbase + r + moff;
            int m1 = gbase + 16 + r + moff;
            out[m0 * COUT + n0] = acc00[r];
            out[m1 * COUT + n0] = acc10[r];
            if (n1 < COUT) {
                out[m0 * COUT + n1] = acc01[r];
                out[m1 * COUT + n1] = acc11[r];
            }
        }
    } else {
#pragma unroll
        for (int r = 0; r < 8; ++r) {
            int m0 = gbase + r + moff;
            int m1 = gbase + 16 + r + moff;
            if (m0 < Npts)               out[m0 * COUT + n0] = acc00[r];
            if (m1 < Npts)               out[m1 * COUT + n0] = acc10[r];
            if (n1 < COUT) {
                if (m0 < Npts)           out[m0 * COUT + n1] = acc01[r];
                if (m1 < Npts)           out[m1 * COUT + n1] = acc11[r];
            }
        }
    }
}

extern "C" void kernel_launch(void* const* d_in, const int* in_sizes, int n_in,
                              void* d_out, int out_size, void* d_ws, size_t ws_size,
                              hipStream_t stream) {
    const float* xyz      = (const float*)d_in[0];
    const float* xy_plane = (const float*)d_in[1];
    const float* xz_plane = (const float*)d_in[2];
    const float* yz_plane = (const float*)d_in[3];
    const float* x_vec    = (const float*)d_in[4];
    const float* y_vec    = (const float*)d_in[5];
    const float* z_vec    = (const float*)d_in[6];
    const float* f_vec    = (const float*)d_in[7];
    const float* xyz_min  = (const float*)d_in[8];
    const float* xyz_max  = (const float*)d_in[9];
    float* out = (float*)d_out;

    // workspace layout: 3 transposed planes + 3 transposed vectors (~19 MB)
    float* ws   = (float*)d_ws;
    float* xy_t = ws;
    float* xz_t = xy_t + PLANE_T_ELEMS;
    float* yz_t = xz_t + PLANE_T_ELEMS;
    float* x_t  = yz_t + PLANE_T_ELEMS;
    float* y_t  = x_t + VEC_T_ELEMS;
    float* z_t  = y_t + VEC_T_ELEMS;

    // pass 1: channel-major staging
    {
        int total = PLANE_T_ELEMS;
        int blks  = (total + 255) / 256;
        transpose_cm_kernel<<<blks, 256, 0, stream>>>(xy_plane, xy_t, PLANE_MN, total);
        transpose_cm_kernel<<<blks, 256, 0, stream>>>(xz_plane, xz_t, PLANE_MN, total);
        transpose_cm_kernel<<<blks, 256, 0, stream>>>(yz_plane, yz_t, PLANE_MN, total);
        int vtotal = VEC_T_ELEMS;
        int vblks  = (vtotal + 255) / 256;
        transpose_cm_kernel<<<vblks, 256, 0, stream>>>(x_vec, x_t, GRID_DIM, vtotal);
        transpose_cm_kernel<<<vblks, 256, 0, stream>>>(y_vec, y_t, GRID_DIM, vtotal);
        transpose_cm_kernel<<<vblks, 256, 0, stream>>>(z_vec, z_t, GRID_DIM, vtotal);
    }

    // pass 2: fused sample + WMMA projection
    const int Npts   = in_sizes[0] / 3;
    const int blocks = (Npts + PTS - 1) / PTS;
    tensorf_fused_kernel<<<blocks, PTS, 0, stream>>>(
        xyz, xy_t, xz_t, yz_t, x_t, y_t, z_t,
        f_vec, xyz_min, xyz_max, out, Npts);
}